// StackLSTM_17918603559185
// MI455X (gfx1250) — compile-verified
//
#include <hip/hip_runtime.h>
#include <math.h>

// Problem constants (reference: L=2, B=32, H=4096)
#define LNUM 2
#define BSZ  32
#define HSZ  4096
#define GSZ  (4 * HSZ)          // 16384 gate dim (i,f,g,o)

#define KC     32               // K-chunk per pipeline stage
#define KCP    36               // padded LDS row stride (words) -> conflict-free ds_load_b64
#define TILE_WORDS (16 * KCP)   // 576 floats per 16xKC tile
#define WAVE_WORDS (4 * TILE_WORDS) // A,B x double buffer = 2304 floats / wave

typedef __attribute__((ext_vector_type(2))) float v2f;
typedef __attribute__((ext_vector_type(4))) float v4f;
typedef __attribute__((ext_vector_type(8))) float v8f;
typedef __attribute__((ext_vector_type(4))) int   v4i;

typedef v4i __attribute__((address_space(1)))* gv4i_ptr;  // global v4i*
typedef v4i __attribute__((address_space(3)))* lv4i_ptr;  // LDS v4i*

#if __has_builtin(__builtin_amdgcn_global_load_async_to_lds_b128)
#define HAVE_ASYNC_LDS 1
#else
#define HAVE_ASYNC_LDS 0
#endif

// -----------------------------------------------------------------------------
// Wait until <= N async (global->LDS) ops outstanding; compiler memory barrier
// keeps the subsequent ds_loads from being hoisted above the wait.
// -----------------------------------------------------------------------------
template <int N>
__device__ __forceinline__ void wait_async_le() {
#if HAVE_ASYNC_LDS
#if __has_builtin(__builtin_amdgcn_s_wait_asynccnt)
    __builtin_amdgcn_s_wait_asynccnt((short)N);
#else
    asm volatile("s_wait_asynccnt %0" :: "i"(N) : "memory");
#endif
#endif
    asm volatile("" ::: "memory");
}

// -----------------------------------------------------------------------------
// Stage one 16 x KC fp32 tile: global (row-major, row stride HSZ) -> LDS
// (row stride KCP). Fully coalesced: per instruction, 8 lanes x 16B cover one
// 128B row, 4 rows/instruction, 4 instructions per tile. Async when available.
// gsrc points at element (row 0, col k) of the tile.
// -----------------------------------------------------------------------------
__device__ __forceinline__ void tile_load(const float* __restrict__ gsrc,
                                          float* lds, int rq, int sub) {
#pragma unroll
    for (int j = 0; j < 4; ++j) {
        const int r = j * 4 + rq;                       // tile row 0..15
        const float* gp = gsrc + (size_t)r * HSZ + sub * 4;
        float* lp = lds + r * KCP + sub * 4;
#if HAVE_ASYNC_LDS
        __builtin_amdgcn_global_load_async_to_lds_b128(
            (gv4i_ptr)gp, (lv4i_ptr)lp, /*imm_offset=*/0, /*cpol=*/0);
#else
        *(v4f*)lp = *(const v4f*)gp;   // sync fallback: b128 load + ds_store_b128
#endif
    }
}

// -----------------------------------------------------------------------------
// 8 chained V_WMMA_F32_16X16X4_F32 over one staged K-chunk.
// Operand layout (ISA 7.12.2): lane needs the float2 at column kk + 2*(lane>=16)
// of its row (A: row = batch m, B: row = gate n since B(k,n) = W[n][k]).
// KCP=36 word stride => 36*r mod 64 hits disjoint bank quads for all 16 rows,
// so the 32-lane ds_load_b64 is bank-conflict-free.
// -----------------------------------------------------------------------------
__device__ __forceinline__ void chunk_mma(const float* aT, const float* bT,
                                          int lrow, int khalf, v8f& acc) {
#pragma unroll
    for (int kk = 0; kk < KC; kk += 4) {
        v2f a = *(const v2f*)(aT + lrow * KCP + kk + 2 * khalf);
        v2f b = *(const v2f*)(bT + lrow * KCP + kk + 2 * khalf);
        acc = __builtin_amdgcn_wmma_f32_16x16x4_f32(
            false, a, false, b, (short)0, acc, false, false);
    }
}

// -----------------------------------------------------------------------------
// One pipelined GEMM accumulation: acc += A[16,H] x B[16,H]^T over K = H.
// Double-buffered async staging: while chunk c's 8 WMMAs run, chunk c+1's
// 8 coalesced b128 global->LDS transfers are in flight (ASYNCcnt <= 8).
// unroll 2 => (c & 1) buffer selects are compile-time: no cndmask pointer
// selection in the steady state, LDS addresses live in hoisted registers.
// -----------------------------------------------------------------------------
__device__ __forceinline__ void gemm_pipelined(
    const float* __restrict__ Ag,    // A tile base: rows m0.., col 0
    const float* __restrict__ Bg,    // B tile base: rows n0.., col 0
    float* aT0, float* aT1, float* bT0, float* bT1,
    int lrow, int khalf, int rq, int sub, v8f& acc)
{
    constexpr int NC = HSZ / KC;     // 128 chunks
    tile_load(Ag, aT0, rq, sub);     // prologue: stage chunk 0
    tile_load(Bg, bT0, rq, sub);
#pragma unroll 2
    for (int c = 0; c < NC; ++c) {
        float* aCur = (c & 1) ? aT1 : aT0;
        float* bCur = (c & 1) ? bT1 : bT0;
        if (c + 1 < NC) {                        // uniform branch (EXEC intact)
            const int off = (c + 1) * KC;        // strength-reduced to ptr incr
            tile_load(Ag + off, (c & 1) ? aT0 : aT1, rq, sub);
            tile_load(Bg + off, (c & 1) ? bT0 : bT1, rq, sub);
            wait_async_le<8>();                  // chunk c landed; c+1 in flight
        } else {
            wait_async_le<0>();                  // drain for final chunk
        }
        chunk_mma(aCur, bCur, lrow, khalf, acc);
    }
}

// -----------------------------------------------------------------------------
// Phase 1: gates[B,4H] = X @ Wih^T + Hp @ Whh^T.
// One wave per 16x16 output tile; 2 M-tiles x 1024 N-tiles = 2048 waves
// (256 blocks x 8 waves). All LDS is wave-private -> no barriers, EXEC all 1s.
// -----------------------------------------------------------------------------
__global__ __launch_bounds__(256) void lstm_gates_wmma(
    const float* __restrict__ X,     // [B, H]   layer input
    const float* __restrict__ Hp,    // [B, H]   prev hidden
    const float* __restrict__ Wih,   // [4H, H]
    const float* __restrict__ Whh,   // [4H, H]
    float* __restrict__ gates)       // [B, 4H]
{
    __shared__ float lds[8 * WAVE_WORDS];   // 8 waves x 9216 B = 72 KB

    const int lane  = threadIdx.x & 31;
    const int wave  = threadIdx.x >> 5;
    const int gw    = blockIdx.x * 8 + wave;   // 0..2047
    const int mtile = gw >> 10;                // 0..1
    const int ntile = gw & 1023;               // 0..1023
    const int m0    = mtile * 16;
    const int n0    = ntile * 16;
    const int lrow  = lane & 15;
    const int khalf = lane >> 4;               // 0 or 1 (K-half for WMMA operands)
    const int rq    = lane >> 3;               // 0..3   (row group for tile fill)
    const int sub   = lane & 7;                // 16B slot within a 128B row

    float* base = lds + wave * WAVE_WORDS;
    float* aT0 = base;
    float* aT1 = base + TILE_WORDS;
    float* bT0 = base + 2 * TILE_WORDS;
    float* bT1 = base + 3 * TILE_WORDS;

    v8f acc = {};
    // x @ Wih^T
    gemm_pipelined(X  + (size_t)m0 * HSZ, Wih + (size_t)n0 * HSZ,
                   aT0, aT1, bT0, bT1, lrow, khalf, rq, sub, acc);
    // h @ Whh^T (same accumulator)
    gemm_pipelined(Hp + (size_t)m0 * HSZ, Whh + (size_t)n0 * HSZ,
                   aT0, aT1, bT0, bT1, lrow, khalf, rq, sub, acc);

    // Write back C/D tile per ISA layout: VGPR r -> row m0 + r + 8*khalf
#pragma unroll
    for (int r = 0; r < 8; ++r) {
        const int row = m0 + r + 8 * khalf;
        const int col = n0 + lrow;
        gates[(size_t)row * GSZ + col] = acc[r];
    }
}

// -----------------------------------------------------------------------------
// Phase 2: elementwise LSTM gating over B*H elements.
//   c_new = sigmoid(f)*c + sigmoid(i)*tanh(g);  h_new = sigmoid(o)*tanh(c_new)
// -----------------------------------------------------------------------------
__device__ __forceinline__ float sigmoidf_(float x) {
    return 1.0f / (1.0f + __expf(-x));
}

__global__ __launch_bounds__(256) void lstm_pointwise(
    const float* __restrict__ gates,  // [B, 4H]
    const float* __restrict__ Cp,     // [B, H]
    const float* __restrict__ bih,    // [4H]
    const float* __restrict__ bhh,    // [4H]
    float* __restrict__ outH,         // [B, H]
    float* __restrict__ outC)         // [B, H]
{
    const int idx = blockIdx.x * blockDim.x + threadIdx.x;  // 0..B*H-1
    const int b = idx >> 12;          // / H (H = 4096 = 2^12)
    const int h = idx & (HSZ - 1);

    const float* g = gates + (size_t)b * GSZ;
    const float gi = g[h]           + bih[h]           + bhh[h];
    const float gf = g[HSZ + h]     + bih[HSZ + h]     + bhh[HSZ + h];
    const float gg = g[2 * HSZ + h] + bih[2 * HSZ + h] + bhh[2 * HSZ + h];
    const float go = g[3 * HSZ + h] + bih[3 * HSZ + h] + bhh[3 * HSZ + h];

    const float c  = Cp[idx];
    const float cn = sigmoidf_(gf) * c + sigmoidf_(gi) * tanhf(gg);
    const float hn = sigmoidf_(go) * tanhf(cn);

    outH[idx] = hn;
    outC[idx] = cn;
}

// -----------------------------------------------------------------------------
// Host-side launch: 2 sequential layers; layer 1's x input is layer 0's next_h
// read straight from d_out (stream-ordered). gates scratch lives in d_ws.
// d_out layout: [2, L, B, H] = next_h[L,B,H] then next_c[L,B,H].
// -----------------------------------------------------------------------------
extern "C" void kernel_launch(void* const* d_in, const int* in_sizes, int n_in,
                              void* d_out, int out_size, void* d_ws, size_t ws_size,
                              hipStream_t stream) {
    const float* inputs = (const float*)d_in[0];  // [B, H]
    const float* prev_h = (const float*)d_in[1];  // [L, B, H]
    const float* prev_c = (const float*)d_in[2];  // [L, B, H]
    const float* W_ih   = (const float*)d_in[3];  // [L, 4H, H]
    const float* W_hh   = (const float*)d_in[4];  // [L, 4H, H]
    const float* b_ih   = (const float*)d_in[5];  // [L, 4H]
    const float* b_hh   = (const float*)d_in[6];  // [L, 4H]

    float* out   = (float*)d_out;
    float* gates = (float*)d_ws;                  // B*4H floats = 2 MB scratch

    const size_t BH = (size_t)BSZ * HSZ;          // 131072
    const size_t WL = (size_t)GSZ * HSZ;          // per-layer weight elems

    const float* x = inputs;
    for (int l = 0; l < LNUM; ++l) {
        lstm_gates_wmma<<<256, 256, 0, stream>>>(
            x, prev_h + (size_t)l * BH,
            W_ih + (size_t)l * WL, W_hh + (size_t)l * WL, gates);

        lstm_pointwise<<<(BSZ * HSZ) / 256, 256, 0, stream>>>(
            gates, prev_c + (size_t)l * BH,
            b_ih + (size_t)l * GSZ, b_hh + (size_t)l * GSZ,
            out + (size_t)l * BH,                 // next_h[l]
            out + (size_t)(LNUM + l) * BH);       // next_c[l]

        x = out + (size_t)l * BH;                 // next layer input = h[l]
    }
}